// HardNegativeMiningLoss_62345745269433
// MI455X (gfx1250) — compile-verified
//
#include <hip/hip_runtime.h>
#include <math.h>

#define B_N   8192
#define D_N   512
#define TEMPI (1.0f / 0.07f)
#define KTOP  16
#define CHUNK 2048
#define SIMP  (CHUNK + 4)   // padded LDS row pitch (floats) -> lane halves hit disjoint banks

typedef __attribute__((ext_vector_type(2))) float v2f;
typedef __attribute__((ext_vector_type(8))) float v8f;

__device__ __forceinline__ float waveReduceSum(float v) {
#pragma unroll
  for (int off = 16; off > 0; off >>= 1) v += __shfl_xor(v, off, 32);
  return v;
}

// ---------------------------------------------------------------------------
// Pass 1: per-row positive stats. One wave per row; positives are sparse, so
// scan labels with ballot and compute only the matching dot products.
// ---------------------------------------------------------------------------
__global__ void pass1_pos_stats(const float* __restrict__ E,
                                const long long* __restrict__ lab,
                                float* __restrict__ posMin,
                                float* __restrict__ posAvg,
                                int* __restrict__ rowValid) {
  const int wave = threadIdx.x >> 5;
  const int lane = threadIdx.x & 31;
  const int i = blockIdx.x * 8 + wave;
  const long long li = lab[i];

  // my 16-float slice of E[i]
  const float4* er = ((const float4*)(E + (size_t)i * D_N)) + lane * 4;
  float4 e0 = er[0], e1 = er[1], e2 = er[2], e3 = er[3];

  float pos_sum = 0.f, pos_min = 1e30f;
  int pos_cnt = 0, same_cnt = 0;

  for (int jb = 0; jb < B_N; jb += 32) {
    long long lj = lab[jb + lane];
    unsigned long long m = __ballot(lj == li);   // wave32: low 32 bits
    same_cnt += __popcll(m);
    while (m) {
      int bit = __builtin_ctzll(m);
      m &= m - 1;
      int j = jb + bit;                          // uniform across wave
      if (j == i) continue;
      const float4* fr = ((const float4*)(E + (size_t)j * D_N)) + lane * 4;
      float4 f0 = fr[0], f1 = fr[1], f2 = fr[2], f3 = fr[3];
      float s = e0.x * f0.x + e0.y * f0.y + e0.z * f0.z + e0.w * f0.w;
      s += e1.x * f1.x + e1.y * f1.y + e1.z * f1.z + e1.w * f1.w;
      s += e2.x * f2.x + e2.y * f2.y + e2.z * f2.z + e2.w * f2.w;
      s += e3.x * f3.x + e3.y * f3.y + e3.z * f3.z + e3.w * f3.w;
      s = waveReduceSum(s);                      // uniform
      pos_sum += s;
      pos_cnt += 1;
      pos_min = fminf(pos_min, s);
    }
  }
  if (lane == 0) {
    int neg_cnt = B_N - same_cnt;                // same_cnt includes self
    posMin[i] = pos_min;                         // +1e30 if no positives (matches ref)
    posAvg[i] = pos_cnt ? (pos_sum / (float)pos_cnt) * TEMPI : 0.f;
    rowValid[i] = (pos_cnt > 0 && neg_cnt > 0) ? 1 : 0;
  }
}

// ---------------------------------------------------------------------------
// Pass 2: 16-row tile per block. WMMA fp32 sim tiles (4 concurrent N-tiles
// per wave for load-latency hiding) -> padded LDS strip -> exact streaming
// top-16 (semi-hard and all-negatives lists) -> per-row loss.
// ---------------------------------------------------------------------------
__global__ void pass2_topk_loss(const float* __restrict__ E,
                                const long long* __restrict__ lab,
                                const float* __restrict__ posMin,
                                const float* __restrict__ posAvg,
                                const int* __restrict__ rowValid,
                                float* __restrict__ lossRow) {
  extern __shared__ char smem[];
  float* simS = (float*)smem;                            // 16 * SIMP floats
  int* labS = (int*)(smem + sizeof(float) * 16 * SIMP);  // CHUNK ints

  __shared__ float topV[16][2][KTOP];
  __shared__ int   topJ[16][2][KTOP];
  __shared__ int   topN[16][2];

  const int tid = threadIdx.x;
  const int wave = tid >> 5, lane = tid & 31;
  const int i0 = blockIdx.x * 16;
  const int lhi = lane >> 4, llo = lane & 15;

  if (tid < 32) topN[tid >> 1][tid & 1] = 0;
  __syncthreads();

  // A-fragment base address (constant across chunks)
  const float* arow = E + (size_t)(i0 + llo) * D_N + 2 * lhi;

  for (int c = 0; c < B_N / CHUNK; ++c) {
    const int jbase = c * CHUNK;

    // stage this chunk's labels as int32
    for (int t = tid; t < CHUNK; t += 256) labS[t] = (int)lab[jbase + t];

    // ---- WMMA phase: wave covers 256 columns = 4 groups of 4 N-tiles ----
    for (int ng = 0; ng < 4; ++ng) {
      const int colLocal0 = wave * 256 + ng * 64;
      const int j0 = jbase + colLocal0;
      const float* brow0 = E + (size_t)(j0 + 0  + llo) * D_N + 2 * lhi;
      const float* brow1 = E + (size_t)(j0 + 16 + llo) * D_N + 2 * lhi;
      const float* brow2 = E + (size_t)(j0 + 32 + llo) * D_N + 2 * lhi;
      const float* brow3 = E + (size_t)(j0 + 48 + llo) * D_N + 2 * lhi;

      v8f acc0 = {0.f, 0.f, 0.f, 0.f, 0.f, 0.f, 0.f, 0.f};
      v8f acc1 = acc0, acc2 = acc0, acc3 = acc0;

#pragma unroll 2
      for (int k0 = 0; k0 < D_N; k0 += 4) {
        // issue all loads first: 1 shared A fragment + 4 independent B frags
        v2f a  = *(const v2f*)(arow + k0);
        v2f b0 = *(const v2f*)(brow0 + k0);
        v2f b1 = *(const v2f*)(brow1 + k0);
        v2f b2 = *(const v2f*)(brow2 + k0);
        v2f b3 = *(const v2f*)(brow3 + k0);
        // 4 independent accumulation chains
        acc0 = __builtin_amdgcn_wmma_f32_16x16x4_f32(
            false, a, false, b0, (short)0, acc0, false, false);
        acc1 = __builtin_amdgcn_wmma_f32_16x16x4_f32(
            false, a, false, b1, (short)0, acc1, false, false);
        acc2 = __builtin_amdgcn_wmma_f32_16x16x4_f32(
            false, a, false, b2, (short)0, acc2, false, false);
        acc3 = __builtin_amdgcn_wmma_f32_16x16x4_f32(
            false, a, false, b3, (short)0, acc3, false, false);
      }

      const int col = colLocal0 + llo;
      const int rowBase = 8 * lhi;
#pragma unroll
      for (int r = 0; r < 8; ++r) {
        float* dst = simS + (r + rowBase) * SIMP + col;
        dst[0]  = acc0[r];
        dst[16] = acc1[r];
        dst[32] = acc2[r];
        dst[48] = acc3[r];
      }
    }
    __syncthreads();

    // ---- selection phase: wave handles rows {wave, wave+8} ----
    for (int rr = 0; rr < 2; ++rr) {
      const int row = wave + rr * 8;
      const int gi = i0 + row;
      const int labI = (int)lab[gi];
      const float pmin = posMin[gi];
      for (int l = 0; l < 2; ++l) {          // l=0: semi-hard, l=1: all negs
        const int nOld = topN[row][l];
        float lastV = INFINITY;
        int lastJ = -1;
        float myV = 0.f;                      // lane kk keeps entry kk
        int myJ = -1;
        int nNew = 0;
        for (int kk = 0; kk < KTOP; ++kk) {
          float bV = -INFINITY;
          int bJ = 0x7fffffff;
          // candidates from this chunk
          for (int idx = lane; idx < CHUNK; idx += 32) {
            float v = simS[row * SIMP + idx];
            int j = jbase + idx;
            bool valid = (labS[idx] != labI) && (l == 1 || v < pmin);
            bool keyLess = (v < lastV) || (v == lastV && j > lastJ);
            bool better = (v > bV) || (v == bV && j < bJ);
            if (valid && keyLess && better) { bV = v; bJ = j; }
          }
          // carried candidates from previous chunks
          if (lane < nOld) {
            float v = topV[row][l][lane];
            int j = topJ[row][l][lane];
            bool keyLess = (v < lastV) || (v == lastV && j > lastJ);
            bool better = (v > bV) || (v == bV && j < bJ);
            if (keyLess && better) { bV = v; bJ = j; }
          }
          // butterfly argmax over (v desc, j asc)
#pragma unroll
          for (int off = 16; off > 0; off >>= 1) {
            float oV = __shfl_xor(bV, off, 32);
            int oJ = __shfl_xor(bJ, off, 32);
            if (oV > bV || (oV == bV && oJ < bJ)) { bV = oV; bJ = oJ; }
          }
          if (bV == -INFINITY) break;        // uniform
          if (lane == kk) { myV = bV; myJ = bJ; }
          lastV = bV; lastJ = bJ;
          nNew = kk + 1;
        }
        if (lane < nNew) { topV[row][l][lane] = myV; topJ[row][l][lane] = myJ; }
        if (lane == 0) topN[row][l] = nNew;
      }
    }
    __syncthreads();   // protect simS/labS reuse next chunk
  }

  // ---- per-row loss: masked logsumexp over the chosen sorted list ----
  for (int rr = 0; rr < 2; ++rr) {
    const int row = wave + rr * 8;
    const int gi = i0 + row;
    const int l = (topN[row][0] > 0) ? 0 : 1;  // semi-hard if any exist
    const int n = topN[row][l];
    float loss = 0.f;
    if (rowValid[gi] && n > 0) {
      const float mx = topV[row][l][0] * TEMPI;        // list sorted desc
      float p = (lane < n) ? expf(topV[row][l][lane] * TEMPI - mx) : 0.f;
      p = waveReduceSum(p);                            // >= 1, log safe
      loss = mx + logf(p) - posAvg[gi];
    }
    if (lane == 0) lossRow[gi] = loss;
  }
}

// ---------------------------------------------------------------------------
// Pass 3: deterministic single-block reduction -> scalar loss.
// ---------------------------------------------------------------------------
__global__ void pass3_reduce(const float* __restrict__ lossRow,
                             const int* __restrict__ rowValid,
                             float* __restrict__ out) {
  __shared__ float sSum[256];
  __shared__ int sCnt[256];
  float s = 0.f;
  int c = 0;
  for (int i = threadIdx.x; i < B_N; i += 256) {
    if (rowValid[i]) { s += lossRow[i]; c += 1; }
  }
  sSum[threadIdx.x] = s;
  sCnt[threadIdx.x] = c;
  __syncthreads();
  for (int off = 128; off > 0; off >>= 1) {
    if (threadIdx.x < off) {
      sSum[threadIdx.x] += sSum[threadIdx.x + off];
      sCnt[threadIdx.x] += sCnt[threadIdx.x + off];
    }
    __syncthreads();
  }
  if (threadIdx.x == 0) out[0] = sSum[0] / (float)(sCnt[0] > 0 ? sCnt[0] : 1);
}

extern "C" void kernel_launch(void* const* d_in, const int* in_sizes, int n_in,
                              void* d_out, int out_size, void* d_ws, size_t ws_size,
                              hipStream_t stream) {
  const float* E = (const float*)d_in[0];
  const long long* lab = (const long long*)d_in[1];

  float* ws = (float*)d_ws;
  float* posMin = ws;                   // [B]
  float* posAvg = ws + B_N;             // [B]
  int* rowValid = (int*)(ws + 2 * B_N); // [B]
  float* lossRow = ws + 3 * B_N;        // [B]

  pass1_pos_stats<<<B_N / 8, 256, 0, stream>>>(E, lab, posMin, posAvg, rowValid);

  size_t smem = sizeof(float) * 16 * SIMP + sizeof(int) * CHUNK;
  pass2_topk_loss<<<B_N / 16, 256, smem, stream>>>(E, lab, posMin, posAvg,
                                                   rowValid, lossRow);

  pass3_reduce<<<1, 256, 0, stream>>>(lossRow, rowValid, (float*)d_out);
}